// CenterHeadTemplate_89223650607526
// MI455X (gfx1250) — compile-verified
//
#include <hip/hip_runtime.h>
#include <hip/hip_bf16.h>
#include <math.h>

// ---------------- problem constants (reference shapes are fixed) -----------
#define NB    32            // batch
#define HH    1024
#define WW    1024
#define HWSZ  (HH * WW)
#define NBINS 2048          // value histogram bins over [0,1)
#define CAP   4096          // candidate capacity per batch
#define TW    128           // tile width  per block
#define TH    32            // tile height per block
#define LW    136           // TW + 4 halo/align cols each side
#define LH    34            // TH + 1 halo row each side
#define SORT_N 4096

// ---------------- workspace layout (4-byte units) --------------------------
#define WS_HIST   0
#define WS_COUNT  (NB * NBINS)          // 65536
#define WS_THRESH (WS_COUNT + NB)       // 65568
#define WS_CVAL   (WS_THRESH + NB)      // 65600
#define WS_CIDX   (WS_CVAL + NB * CAP)  // 196672
// total = 327744 floats ~= 1.31 MB

#if __has_builtin(__builtin_amdgcn_tensor_load_to_lds)
#define USE_TDM 1
#endif
#if __has_builtin(__builtin_amdgcn_global_load_async_to_lds_b128)
#define USE_ASYNC 1
#endif

typedef unsigned int v4u __attribute__((ext_vector_type(4)));
typedef int          v4i __attribute__((ext_vector_type(4)));
typedef int          v8i __attribute__((ext_vector_type(8)));

__device__ __forceinline__ unsigned lds_byte_addr(const void* p) {
  return (unsigned)(unsigned long long)(__attribute__((address_space(3))) const void*)p;
}

// Stage a (LH x LW) fp32 tile covering image rows [y0-1, y0+TH+1) and
// columns [x0-4, x0+TW+4) into LDS. Interior tiles: one TDM descriptor.
// Border tiles: -inf prefill + per-chunk async b128 copies of valid chunks.
__device__ void stage_tile(float (*tile)[LW], const float* __restrict__ src,
                           int x0, int y0) {
  const int tid = threadIdx.x;
  bool interior = (x0 >= 4) && (x0 + TW + 4 <= WW) && (y0 >= 1) && (y0 + TH + 1 <= HH);

#ifdef USE_TDM
  if (interior) {
    if (tid < 32) {  // wave 0 only issues the tensor DMA (EXEC-independent op)
      unsigned long long ga =
          (unsigned long long)(src + (size_t)(y0 - 1) * WW + (x0 - 4));
      unsigned lb = lds_byte_addr(&tile[0][0]);
      // D# group0: count=1 | lds_addr | global_addr[56:0] | type=2
      v4u g0 = { 1u,
                 lb,
                 (unsigned)(ga & 0xFFFFFFFFu),
                 (unsigned)((ga >> 32) & 0x1FFFFFFu) | (2u << 30) };
      // D# group1: data_size=4B; tensor_dim0/1 huge (no clipping, interior);
      // tile 136 x 34; tensor_dim0_stride = WW
      v8i g1 = { (int)(2u << 16),                       // data_size=2 (4B)
                 (int)0xFFFF0000u,                      // tensor_dim0 lo16<<16
                 (int)0xFFFF7FFFu,                      // td0 hi16 | td1 lo16<<16
                 (int)((136u << 16) | 0x7FFFu),         // td1 hi16 | tile_dim0
                 34,                                    // tile_dim1 (tile_dim2=0)
                 WW,                                    // dim0_stride lo32
                 0, 0 };
      v4i g2 = { 0, 0, 0, 0 };
      v4i g3 = { 0, 0, 0, 0 };
      v8i g4 = { 0, 0, 0, 0, 0, 0, 0, 0 };
      __builtin_amdgcn_tensor_load_to_lds(g0, g1, g2, g3, g4, 0);
#if __has_builtin(__builtin_amdgcn_s_wait_tensorcnt)
      __builtin_amdgcn_s_wait_tensorcnt(0);
#else
      asm volatile("s_wait_tensorcnt 0x0" ::: "memory");
#endif
    }
    __syncthreads();
    return;
  }
#endif

  // Border path: -inf fill (max-pool padding semantics), then valid chunks.
  for (int i = tid; i < LH * LW; i += blockDim.x)
    tile[i / LW][i % LW] = -INFINITY;
  __syncthreads();

  const int NCH = LH * (LW / 4);  // 16B chunks
  for (int c = tid; c < NCH; c += blockDim.x) {
    int r  = c / (LW / 4);
    int j  = c % (LW / 4);
    int gy = y0 - 1 + r;
    int gx = x0 - 4 + 4 * j;    // 16B aligned; fully in or fully out of row
    if ((unsigned)gy < (unsigned)HH && (unsigned)gx < (unsigned)WW) {
      const float* gp = src + (size_t)gy * WW + gx;
#ifdef USE_ASYNC
      __builtin_amdgcn_global_load_async_to_lds_b128(
          (__attribute__((address_space(1))) v4i*)gp,
          (__attribute__((address_space(3))) v4i*)&tile[r][4 * j], 0, 0);
#else
      float4 d = *(const float4*)gp;
      *(float4*)&tile[r][4 * j] = d;
#endif
    }
  }
#ifdef USE_ASYNC
#if __has_builtin(__builtin_amdgcn_s_wait_asynccnt)
  __builtin_amdgcn_s_wait_asynccnt(0);
#else
  asm volatile("s_wait_asynccnt 0x0" ::: "memory");
#endif
#endif
  __syncthreads();
}

// MODE 0: build per-batch value histogram of NMS peaks.
// MODE 1: compact peaks with bin >= threshold into candidate buffers.
template <int MODE>
__global__ __launch_bounds__(256) void nms_pass(const float* __restrict__ scores,
                                                float* __restrict__ ws,
                                                const int* __restrict__ kp) {
  __shared__ float tile[LH][LW];
  const int b  = blockIdx.z;
  const int y0 = blockIdx.y * TH;
  const int x0 = blockIdx.x * TW;
  const float* src = scores + (size_t)b * HWSZ;
  stage_tile(tile, src, x0, y0);

  const int tid  = threadIdx.x;
  const int c    = tid & (TW - 1);   // column within tile
  const int half = tid >> 7;         // 0: rows 0..15, 1: rows 16..31
  const int lr   = half * 16;        // LDS row of first window top
  const int lc   = c + 4;

  int*       hist  = (int*)ws + WS_HIST + b * NBINS;
  int*       count = (int*)ws + WS_COUNT;
  const int* thr   = (const int*)ws + WS_THRESH;
  float*     cval  = ws + WS_CVAL + (size_t)b * CAP;
  int*       cidx  = (int*)ws + WS_CIDX + (size_t)b * CAP;
  const int  tb    = (MODE == 1) ? thr[b] : 0;

  float rm0 = fmaxf(fmaxf(tile[lr][lc - 1], tile[lr][lc]), tile[lr][lc + 1]);
  float rm1 = fmaxf(fmaxf(tile[lr + 1][lc - 1], tile[lr + 1][lc]), tile[lr + 1][lc + 1]);
#pragma unroll 4
  for (int i = 0; i < 16; ++i) {
    const int rr = lr + i + 2;
    float rm2 = fmaxf(fmaxf(tile[rr][lc - 1], tile[rr][lc]), tile[rr][lc + 1]);
    float hm  = tile[lr + i + 1][lc];
    float m   = fmaxf(fmaxf(rm0, rm1), rm2);
    if (hm > 0.f && m == hm) {  // peak survives NMS (keep == 1)
      int bin = (int)(hm * (float)NBINS);
      bin = bin < 0 ? 0 : (bin >= NBINS ? NBINS - 1 : bin);
      if (MODE == 0) {
        atomicAdd(&hist[bin], 1);
      } else if (bin >= tb) {
        int pos = atomicAdd(&count[b], 1);
        if (pos < CAP) {
          cval[pos] = hm;
          cidx[pos] = (y0 + half * 16 + i) * WW + (x0 + c);
        }
      }
    }
    rm0 = rm1;
    rm1 = rm2;
  }
  (void)kp;
}

// Per-batch threshold bin: largest bin t s.t. count(value-bin >= t) >= K.
__global__ void select_threshold(float* __restrict__ ws, const int* __restrict__ kp) {
  const int b = blockIdx.x;
  if (threadIdx.x != 0) return;
  const int* hist = (const int*)ws + WS_HIST + b * NBINS;
  const int  K    = *kp;
  int cum = 0, t = 0;
  for (int bin = NBINS - 1; bin >= 0; --bin) {
    cum += hist[bin];
    if (cum >= K) { t = bin; break; }
  }
  ((int*)ws)[WS_THRESH + b] = t;
}

// One block per batch: exact sorted top-K of the candidates via LDS bitonic
// sort (descending value, ties by ascending index = lax.top_k order), then
// emit all six outputs including the 5-float regression gather.
__global__ __launch_bounds__(1024) void final_topk(const float* __restrict__ reg,
                                                   float* __restrict__ ws,
                                                   const int* __restrict__ kp,
                                                   float* __restrict__ out) {
  __shared__ float sval[SORT_N];
  __shared__ int   sidx[SORT_N];
  const int b   = blockIdx.x;
  const int tid = threadIdx.x;
  int n = ((int*)ws)[WS_COUNT + b];
  if (n > CAP) n = CAP;
  const float* cval = ws + WS_CVAL + (size_t)b * CAP;
  const int*   cidx = (const int*)ws + WS_CIDX + (size_t)b * CAP;

  for (int i = tid; i < SORT_N; i += 1024) {
    if (i < n) { sval[i] = cval[i]; sidx[i] = cidx[i]; }
    else       { sval[i] = -INFINITY; sidx[i] = 0x7FFFFFFF; }
  }

  for (int k = 2; k <= SORT_N; k <<= 1) {
    for (int j = k >> 1; j > 0; j >>= 1) {
      __syncthreads();
      for (int i = tid; i < SORT_N; i += 1024) {
        const int l = i ^ j;
        if (l > i) {
          float vi = sval[i], vl = sval[l];
          int   ii = sidx[i], il = sidx[l];
          bool before = (vi > vl) || (vi == vl && ii < il);
          bool desc   = ((i & k) == 0);
          if (desc ? !before : before) {
            sval[i] = vl; sval[l] = vi;
            sidx[i] = il; sidx[l] = ii;
          }
        }
      }
    }
  }
  __syncthreads();

  const int    K  = *kp;
  const size_t BK = (size_t)gridDim.x * (size_t)K;
  float* o_score = out;            // [B,K]
  float* o_regs  = out + BK;       // [B,K,5]
  float* o_ys    = out + 6 * BK;   // [B,K]
  float* o_xs    = out + 7 * BK;   // [B,K]
  float* o_cls   = out + 8 * BK;   // [B,K] (C==1 -> all zero)
  float* o_ind   = out + 9 * BK;   // [B,K]

  for (int k2 = tid; k2 < K; k2 += 1024) {
    float v = 0.f;
    int   id = 0;
    if (k2 < n) { v = sval[k2]; id = sidx[k2]; }
    const size_t o = (size_t)b * K + k2;
    o_score[o] = v;
    o_ys[o]    = (float)(id / WW);
    o_xs[o]    = (float)(id % WW);
    o_cls[o]   = 0.f;
    o_ind[o]   = (float)id;
    const float* rf = reg + ((size_t)b * HWSZ + (size_t)id) * 5;
    float*       ro = o_regs + o * 5;
#pragma unroll
    for (int j2 = 0; j2 < 5; ++j2) ro[j2] = rf[j2];
  }
}

extern "C" void kernel_launch(void* const* d_in, const int* in_sizes, int n_in,
                              void* d_out, int out_size, void* d_ws, size_t ws_size,
                              hipStream_t stream) {
  (void)in_sizes; (void)n_in; (void)out_size; (void)ws_size;
  const float* scores = (const float*)d_in[0];  // [32,1,1024,1024] f32
  const float* reg    = (const float*)d_in[1];  // [32,1024,1024,5] f32
  const int*   kp     = (const int*)d_in[2];    // K (==500)
  float* ws  = (float*)d_ws;
  float* out = (float*)d_out;

  // zero hist + per-batch candidate counters (capture-safe memset node)
  (void)hipMemsetAsync(d_ws, 0, (size_t)(NB * NBINS + NB) * sizeof(int), stream);

  dim3 grid(WW / TW, HH / TH, NB);  // (8, 32, 32)
  nms_pass<0><<<grid, 256, 0, stream>>>(scores, ws, kp);
  select_threshold<<<NB, 32, 0, stream>>>(ws, kp);
  nms_pass<1><<<grid, 256, 0, stream>>>(scores, ws, kp);
  final_topk<<<NB, 1024, 0, stream>>>(reg, ws, kp, out);
}